// GraphSAGE_56813827392143
// MI455X (gfx1250) — compile-verified
//
#include <hip/hip_runtime.h>

// ---------------------------------------------------------------------------
// GraphSAGE 2-layer forward for MI455X (gfx1250, wave32).
//  - edge aggregation: L2-resident f32 atomics (x/agg/h all fit in 192MB L2)
//  - node update: V_WMMA_F32_16X16X4_F32 (full f32 precision, matches ref),
//    4-way row register blocking: each wave holds 4 accumulators and reuses
//    each weight B-fragment across 4 row tiles (64 nodes per block).
// ---------------------------------------------------------------------------

typedef __attribute__((ext_vector_type(2))) float v2f;
typedef __attribute__((ext_vector_type(8))) float v8f;

#define DFEAT 128
#define ROW_TILES 4
#define ROWS_PER_BLOCK (ROW_TILES * 16)  // 64 nodes per block
#define SMPAD 132  // 128 + 4 floats pad -> lanes land 4 banks apart, no conflicts

// ---------------------------------------------------------------- zero fill
__global__ __launch_bounds__(256) void sage_zero(float* __restrict__ p, int n) {
  int t = blockIdx.x * blockDim.x + threadIdx.x;
  if (t < n) p[t] = 0.0f;
}

// ------------------------------------------------------------ degree counts
__global__ __launch_bounds__(256) void sage_degree(const long long* __restrict__ dst,
                                                   float* __restrict__ cnt,
                                                   int n_edges) {
  int t = blockIdx.x * blockDim.x + threadIdx.x;
  if (t < n_edges) {
    unsafeAtomicAdd(&cnt[(int)dst[t]], 1.0f);
  }
}

// ------------------------------------------- edge scatter: one wave per edge
// lane l handles features [4l, 4l+4): float4 gather from src row, 4 f32
// atomics into dst row. Feature tables are L2-resident (25.6MB each).
__global__ __launch_bounds__(256) void sage_scatter(const float* __restrict__ feat,
                                                    const long long* __restrict__ src,
                                                    const long long* __restrict__ dst,
                                                    float* __restrict__ agg,
                                                    int n_edges) {
  int t = blockIdx.x * blockDim.x + threadIdx.x;
  int e = t >> 5;
  int lane = t & 31;
  if (e >= n_edges) return;
  int s = (int)src[e];
  int d = (int)dst[e];
  const float4 v = *(const float4*)(feat + (size_t)s * DFEAT + lane * 4);
  float* a = agg + (size_t)d * DFEAT + lane * 4;
  unsafeAtomicAdd(a + 0, v.x);
  unsafeAtomicAdd(a + 1, v.y);
  unsafeAtomicAdd(a + 2, v.z);
  unsafeAtomicAdd(a + 3, v.w);
}

// --------------------------------------------------- mean = agg / max(cnt,1)
__global__ __launch_bounds__(256) void sage_mean(float* __restrict__ agg,
                                                 const float* __restrict__ cnt,
                                                 int n_nodes) {
  int t = blockIdx.x * blockDim.x + threadIdx.x;
  if (t < n_nodes * DFEAT) {
    float c = fmaxf(cnt[t >> 7], 1.0f);
    agg[t] = agg[t] / c;
  }
}

// ---------------------------------------------------------------------------
// Fused SAGE node update over a 64-node block:
//   out[64 x 128] = act( mean@W_l + bias + root@W_r )
// Block = 256 threads = 8 waves; wave w owns output columns [16w, 16w+16) for
// all 4 row tiles. Per k-step of 4: load one B-fragment pair (W_l, W_r), run
// 8 v_wmma_f32_16x16x4_f32 (2 per row tile) -> 256 WMMAs per wave.
//
// Fragment mapping (ISA 7.12.2, 32-bit, wave32):
//   A 16x4 : lane -> M = lane&15, K-pair = lane>>4 ; v2f = {K, K+1}
//   B 4x16 : lane -> N = lane&15, K-pair = lane>>4 ; v2f = {K, K+1}
//   C/D    : elem i -> M = i + 8*(lane>>4), N = lane&15
// ---------------------------------------------------------------------------
__global__ __launch_bounds__(256) void sage_gemm(const float* __restrict__ meanf,
                                                 const float* __restrict__ rootf,
                                                 const float* __restrict__ Wl,
                                                 const float* __restrict__ Wr,
                                                 const float* __restrict__ bias,
                                                 float* __restrict__ out,
                                                 int n_nodes, int do_relu) {
  __shared__ float smM[ROWS_PER_BLOCK * SMPAD];
  __shared__ float smR[ROWS_PER_BLOCK * SMPAD];

  const int tid = threadIdx.x;
  const int node_base = blockIdx.x * ROWS_PER_BLOCK;

  // ---- cooperative stage: each thread moves 32 floats per matrix (8x float4)
  {
    int row = tid >> 2;            // 0..63
    int colb = (tid & 3) << 5;     // 0,32,64,96
    int grow = node_base + row;
    float* dm = &smM[row * SMPAD + colb];
    float* dr = &smR[row * SMPAD + colb];
    if (grow < n_nodes) {
      const float4* gm = (const float4*)(meanf + (size_t)grow * DFEAT + colb);
      const float4* gr = (const float4*)(rootf + (size_t)grow * DFEAT + colb);
#pragma unroll
      for (int j = 0; j < 8; ++j) {
        float4 mv = gm[j];
        float4 rv = gr[j];
        dm[4 * j + 0] = mv.x; dm[4 * j + 1] = mv.y;
        dm[4 * j + 2] = mv.z; dm[4 * j + 3] = mv.w;
        dr[4 * j + 0] = rv.x; dr[4 * j + 1] = rv.y;
        dr[4 * j + 2] = rv.z; dr[4 * j + 3] = rv.w;
      }
    } else {
#pragma unroll
      for (int j = 0; j < 32; ++j) { dm[j] = 0.0f; dr[j] = 0.0f; }
    }
  }
  __syncthreads();

  const int wave = tid >> 5;            // 0..7 -> 16-column tile
  const int lane = tid & 31;
  const int m = lane & 15;              // A row / B,D column index
  const int khalf = lane >> 4;          // K-pair selector
  const int colN = (wave << 4) + m;     // output column
  const int krow = khalf << 1;

  v8f acc[ROW_TILES];
  {
    float bv = bias[colN];
#pragma unroll
    for (int rt = 0; rt < ROW_TILES; ++rt)
#pragma unroll
      for (int i = 0; i < 8; ++i) acc[rt][i] = bv;
  }

#pragma unroll 4
  for (int kb = 0; kb < DFEAT; kb += 4) {
    const int k0 = kb + krow;
    v2f bl, br;
    bl.x = Wl[(size_t)k0 * DFEAT + colN];
    bl.y = Wl[(size_t)(k0 + 1) * DFEAT + colN];
    br.x = Wr[(size_t)k0 * DFEAT + colN];
    br.y = Wr[(size_t)(k0 + 1) * DFEAT + colN];
#pragma unroll
    for (int rt = 0; rt < ROW_TILES; ++rt) {
      const int arow = (rt << 4) + m;
      v2f am = *(const v2f*)&smM[arow * SMPAD + k0];
      v2f ar = *(const v2f*)&smR[arow * SMPAD + k0];
      acc[rt] = __builtin_amdgcn_wmma_f32_16x16x4_f32(false, am, false, bl,
                                                      (short)0, acc[rt],
                                                      false, false);
      acc[rt] = __builtin_amdgcn_wmma_f32_16x16x4_f32(false, ar, false, br,
                                                      (short)0, acc[rt],
                                                      false, false);
    }
  }

#pragma unroll
  for (int rt = 0; rt < ROW_TILES; ++rt) {
    if (do_relu) {
#pragma unroll
      for (int i = 0; i < 8; ++i) acc[rt][i] = fmaxf(acc[rt][i], 0.0f);
    }
    const int rbase = node_base + (rt << 4) + (khalf << 3);
#pragma unroll
    for (int i = 0; i < 8; ++i) {
      int r = rbase + i;
      if (r < n_nodes) out[(size_t)r * DFEAT + colN] = acc[rt][i];
    }
  }
}

// ---------------------------------------------------------------------------
extern "C" void kernel_launch(void* const* d_in, const int* in_sizes, int n_in,
                              void* d_out, int out_size, void* d_ws, size_t ws_size,
                              hipStream_t stream) {
  const float*     x    = (const float*)d_in[0];
  const long long* edge = (const long long*)d_in[1];  // int64 [2, E]
  const float*     W1l  = (const float*)d_in[2];
  const float*     b1   = (const float*)d_in[3];
  const float*     W1r  = (const float*)d_in[4];
  const float*     W2l  = (const float*)d_in[5];
  const float*     b2   = (const float*)d_in[6];
  const float*     W2r  = (const float*)d_in[7];

  const int n_nodes = in_sizes[0] / DFEAT;
  const int n_edges = in_sizes[1] / 2;
  const long long* src = edge;
  const long long* dst = edge + n_edges;

  // workspace: agg[n*128] | cnt[n] | h[n*128]
  float* agg = (float*)d_ws;
  float* cnt = agg + (size_t)n_nodes * DFEAT;
  float* h   = cnt + n_nodes;

  const int feat_elems = n_nodes * DFEAT;
  const int gemm_blocks = (n_nodes + ROWS_PER_BLOCK - 1) / ROWS_PER_BLOCK;
  const int scat_blocks = (n_edges * 32 + 255) / 256;

  // ----- layer 1 -----
  sage_zero<<<(feat_elems + n_nodes + 255) / 256, 256, 0, stream>>>(
      agg, feat_elems + n_nodes);
  sage_degree<<<(n_edges + 255) / 256, 256, 0, stream>>>(dst, cnt, n_edges);
  sage_scatter<<<scat_blocks, 256, 0, stream>>>(x, src, dst, agg, n_edges);
  sage_mean<<<(feat_elems + 255) / 256, 256, 0, stream>>>(agg, cnt, n_nodes);
  sage_gemm<<<gemm_blocks, 256, 0, stream>>>(agg, x, W1l, W1r, b1, h, n_nodes, 1);

  // ----- layer 2 -----
  sage_zero<<<(feat_elems + 255) / 256, 256, 0, stream>>>(agg, feat_elems);
  sage_scatter<<<scat_blocks, 256, 0, stream>>>(h, src, dst, agg, n_edges);
  sage_mean<<<(feat_elems + 255) / 256, 256, 0, stream>>>(agg, cnt, n_nodes);
  sage_gemm<<<gemm_blocks, 256, 0, stream>>>(agg, h, W2l, W2r, b2,
                                             (float*)d_out, n_nodes, 0);
}